// EnhancedGNNModelWithMLP_33114197852244
// MI455X (gfx1250) — compile-verified
//
#include <hip/hip_runtime.h>

typedef float v2f __attribute__((ext_vector_type(2)));
typedef float v8f __attribute__((ext_vector_type(8)));

#define NNODES 500022
#define NEDGES 2500000
#define HID    128
#define EPSN   1e-5f
#define MLPCH  65536   // MLP row chunk

// ---------------------------------------------------------------------------
// degree count (float) per destination node
__global__ void count_edges(const int* __restrict__ dst, float* __restrict__ cnt, int E) {
    int e = blockIdx.x * blockDim.x + threadIdx.x;
    if (e < E) atomicAdd(&cnt[dst[e]], 1.0f);
}

__global__ void finalize_inv(const float* __restrict__ cnt, float* __restrict__ inv, int n) {
    int i = blockIdx.x * blockDim.x + threadIdx.x;
    if (i < n) inv[i] = 1.0f / fmaxf(cnt[i], 1.0f);
}

// layer-0 aggregation: scalar feature scatter-add
__global__ void scatter_scalar(const float* __restrict__ x, const int* __restrict__ src,
                               const int* __restrict__ dst, float* __restrict__ agg, int E) {
    int e = blockIdx.x * blockDim.x + threadIdx.x;
    if (e < E) atomicAdd(&agg[dst[e]], x[src[e]]);
}

// layers 1..4 aggregation: 128-wide scatter-add, 32 threads/edge, float4 per thread
__global__ void scatter_add128(const float* __restrict__ h, const int* __restrict__ src,
                               const int* __restrict__ dst, float* __restrict__ agg, int E) {
    long long tid = (long long)blockIdx.x * blockDim.x + threadIdx.x;
    long long e   = tid >> 5;
    if (e >= E) return;
    int f = (int)(tid & 31) << 2;
    int s = src[e], d = dst[e];
    const float4 v = *(const float4*)(h + (size_t)s * HID + f);
    float* p = agg + (size_t)d * HID + f;
    atomicAdd(p + 0, v.x); atomicAdd(p + 1, v.y);
    atomicAdd(p + 2, v.z); atomicAdd(p + 3, v.w);
}

// conv0 (fan_in = 1): pre = mean*Wl[j] + bl[j] + x*Wr[j]
__global__ void conv0_kernel(const float* __restrict__ x, const float* __restrict__ agg,
                             const float* __restrict__ inv, const float* __restrict__ Wl,
                             const float* __restrict__ bl, const float* __restrict__ Wr,
                             float* __restrict__ pre, int nrows) {
    long long tid = (long long)blockIdx.x * blockDim.x + threadIdx.x;
    long long n = tid >> 7;
    if (n >= nrows) return;
    int j = (int)(tid & 127);
    float mean = agg[n] * inv[n];
    pre[tid] = mean * Wl[j] + bl[j] + x[n] * Wr[j];
}

// ---------------------------------------------------------------------------
// fp32 WMMA GEMM:  C[nrows,FOUT] = (rowscale.*A)[nrows,K] @ W[K,FOUT]
//                                   (+ A2 @ W2)  (+ bias)  (opt. ReLU)
// Template flags -> no divergent branches anywhere near the WMMAs (full EXEC).
// A tile (16 rows x K) staged in LDS once per block (pad +2 floats/row so the
// 16 lanes of a fragment ds_load_b64 hit 16 distinct banks). 8 waves sweep FOUT.
template<bool DUAL, bool RELU, bool SCALE, int K, int FOUT>
__global__ __launch_bounds__(256) void wmma_gemm(
        const float* __restrict__ A, const float* __restrict__ rowscale,
        const float* __restrict__ W,
        const float* __restrict__ A2, const float* __restrict__ W2,
        const float* __restrict__ bias, float* __restrict__ C, int nrows)
{
    constexpr int KP = K + 2;                       // padded LDS row (bank skew)
    __shared__ float sA[(DUAL ? 2 : 1) * 16 * KP];

    const int m0 = blockIdx.x << 4;
    if (m0 >= nrows) return;                        // block-uniform

    // ---- stage A (and A2) tile into LDS, rowscale folded in ----
    for (int i = threadIdx.x; i < 16 * K; i += 256) {
        int r = i / K, c = i - r * K;               // coalesced along c
        int row  = m0 + r;
        int rowc = row < nrows ? row : (nrows - 1); // tail clamp (rows never stored)
        float v = A[(size_t)rowc * K + c];
        if (SCALE) v *= rowscale[rowc];
        sA[r * KP + c] = v;
        if (DUAL) sA[16 * KP + r * KP + c] = A2[(size_t)rowc * K + c];
    }
    __syncthreads();

    const int lane = threadIdx.x & 31;
    const int wave = threadIdx.x >> 5;
    const int lh   = lane >> 4;                     // lane half (0/1)
    const int ll   = lane & 15;

    constexpr int NCT = FOUT >> 4;
#pragma unroll
    for (int ct = wave; ct < NCT; ct += 8) {        // wave-uniform (compile-time trip)
        const int n0  = ct << 4;
        const int col = n0 + ll;
        v8f acc = {};
        const float* pa  = &sA[ll * KP + (lh << 1)];
        const float* pa2 = &sA[16 * KP + ll * KP + (lh << 1)];
        const float* pw  = W + (size_t)(lh << 1) * FOUT + col;
        const float* pw2 = DUAL ? (W2 + (size_t)(lh << 1) * FOUT + col) : nullptr;
#pragma unroll 8
        for (int k = 0; k < K; k += 4) {
            v2f a = *(const v2f*)pa;                // ds_load_b64, conflict-free
            v2f b; b.x = pw[0]; b.y = pw[FOUT];     // coalesced across lanes
            acc = __builtin_amdgcn_wmma_f32_16x16x4_f32(
                      false, a, false, b, (short)0, acc, false, false);
            if constexpr (DUAL) {
                v2f a2 = *(const v2f*)pa2;
                v2f b2; b2.x = pw2[0]; b2.y = pw2[FOUT];
                acc = __builtin_amdgcn_wmma_f32_16x16x4_f32(
                          false, a2, false, b2, (short)0, acc, false, false);
                pa2 += 4; pw2 += 4 * FOUT;
            }
            pa += 4; pw += 4 * FOUT;
        }
        const float bv = bias[col];
#pragma unroll
        for (int v = 0; v < 8; v++) {               // C layout: row = m0 + v + 8*lh
            int row = m0 + v + (lh << 3);
            float val = acc[v] + bv;
            if (RELU) val = fmaxf(val, 0.0f);
            if (row < nrows) C[(size_t)row * FOUT + col] = val;
        }
    }
}

// ---------------------------------------------------------------------------
// GraphNorm pass 1: per-column sum and sum-of-squares (128 columns)
__global__ void colstats128(const float* __restrict__ X, int nrows, float* __restrict__ stats) {
    int j = threadIdx.x;                                   // 128 threads/block
    float s = 0.0f, s2 = 0.0f;
    for (int r = blockIdx.x; r < nrows; r += gridDim.x) {
        float v = X[(size_t)r * HID + j];
        s += v; s2 += v * v;
    }
    atomicAdd(&stats[j], s);
    atomicAdd(&stats[HID + j], s2);
}

// GraphNorm pass 2 + ReLU:  y = relu(gamma*(x - a*mu)*rsqrt(var+eps) + beta)
// var = E[x^2] - 2*a*mu*mu + a*a*mu*mu  (exact expansion of mean((x-a*mu)^2))
__global__ void graphnorm_relu(const float* __restrict__ X, const float* __restrict__ stats,
                               const float* __restrict__ gamma, const float* __restrict__ beta,
                               const float* __restrict__ alpha, float* __restrict__ Y, int nrows) {
    long long tid = (long long)blockIdx.x * blockDim.x + threadIdx.x;
    if (tid >= (long long)nrows * HID) return;
    int j = (int)(tid & 127);
    float invN = 1.0f / (float)nrows;
    float mu  = stats[j] * invN;
    float ex2 = stats[HID + j] * invN;
    float a   = alpha[j];
    float var = ex2 - 2.0f * a * mu * mu + a * a * mu * mu;
    float xc  = X[tid] - a * mu;
    float y   = gamma[j] * xc * rsqrtf(var + EPSN) + beta[j];
    Y[tid] = fmaxf(y, 0.0f);
}

// tiny final MLP layer (oc <= 7): plain dot products
__global__ void mlp_final(const float* __restrict__ X, const float* __restrict__ W3,
                          const float* __restrict__ b3, float* __restrict__ out,
                          int rows, int oc) {
    long long tid = (long long)blockIdx.x * blockDim.x + threadIdx.x;
    if (tid >= (long long)rows * oc) return;
    int r = (int)(tid / oc), j = (int)(tid - (long long)r * oc);
    const float* xr = X + (size_t)r * HID;
    float s = b3[j];
    for (int k = 0; k < HID; k++) s += xr[k] * W3[(size_t)k * oc + j];
    out[tid] = s;
}

// ---------------------------------------------------------------------------
extern "C" void kernel_launch(void* const* d_in, const int* in_sizes, int n_in,
                              void* d_out, int out_size, void* d_ws, size_t ws_size,
                              hipStream_t stream)
{
    const int N = NNODES;
    const int E = NEDGES;

    // ---- input pointer mapping (setup_inputs insertion order) ----
    const float* x   = (const float*)d_in[0];
    const int*   ei  = (const int*)d_in[1];
    const int*   src = ei;
    const int*   dst = ei + E;
    int idx = 2;
    const float *Wl[5], *bl[5], *Wr[5];
    for (int i = 0; i < 5; i++) {
        Wl[i] = (const float*)d_in[idx++];
        bl[i] = (const float*)d_in[idx++];
        Wr[i] = (const float*)d_in[idx++];
    }
    const float *gma[5], *bta[5], *alp[5];
    for (int i = 0; i < 5; i++) {
        gma[i] = (const float*)d_in[idx++];
        bta[i] = (const float*)d_in[idx++];
        alp[i] = (const float*)d_in[idx++];
    }
    const float *W1[5], *b1[5], *W2[5], *b2[5], *W3[5], *b3[5];
    for (int h = 0; h < 5; h++) {
        W1[h] = (const float*)d_in[idx++];  b1[h] = (const float*)d_in[idx++];
        W2[h] = (const float*)d_in[idx++];  b2[h] = (const float*)d_in[idx++];
        W3[h] = (const float*)d_in[idx++];  b3[h] = (const float*)d_in[idx++];
    }
    const int    oc[5]      = {7, 2, 5, 4, 4};
    const size_t headoff[5] = {0, 7ull * N, 9ull * N, 14ull * N, 18ull * N};

    // ---- workspace carve (256B aligned) ----
    char* ws = (char*)d_ws;
    size_t off = 0;
    auto carve = [&](size_t bytes) -> float* {
        float* p = (float*)(ws + off);
        off += (bytes + 255) & ~(size_t)255;
        return p;
    };
    float* agg   = carve((size_t)N * HID * 4);   // mean-agg accumulator
    float* bufA  = carve((size_t)N * HID * 4);   // h (post-norm activations)
    float* bufB  = carve((size_t)N * HID * 4);   // conv pre-activations
    float* cnt   = carve((size_t)N * 4);
    float* inv   = carve((size_t)N * 4);
    float* stats = carve(2 * HID * 4);
    float* t1    = carve((size_t)MLPCH * 256 * 4);
    float* t2    = carve((size_t)MLPCH * HID * 4);

    const long long NH = (long long)N * HID;
    const int gridEl   = (int)((NH + 255) / 256);           // N*128 elementwise
    const int gridE    = (E + 255) / 256;
    const int gridE32  = (int)(((long long)E * 32 + 255) / 256);
    const int gridRowT = (N + 15) / 16;                     // 16-row GEMM tiles

    // ---- degrees ----
    hipMemsetAsync(cnt, 0, (size_t)N * 4, stream);
    hipMemsetAsync(agg, 0, (size_t)N * HID * 4, stream);
    count_edges<<<gridE, 256, 0, stream>>>(dst, cnt, E);
    finalize_inv<<<(N + 255) / 256, 256, 0, stream>>>(cnt, inv, N);

    // ---- layer 0 (fan_in = 1) ----
    scatter_scalar<<<gridE, 256, 0, stream>>>(x, src, dst, agg, E);
    conv0_kernel<<<gridEl, 256, 0, stream>>>(x, agg, inv, Wl[0], bl[0], Wr[0], bufB, N);
    hipMemsetAsync(stats, 0, 2 * HID * 4, stream);
    colstats128<<<1024, 128, 0, stream>>>(bufB, N, stats);
    graphnorm_relu<<<gridEl, 256, 0, stream>>>(bufB, stats, gma[0], bta[0], alp[0], bufA, N);

    // ---- layers 1..4 ----
    for (int L = 1; L < 5; L++) {
        hipMemsetAsync(agg, 0, (size_t)N * HID * 4, stream);
        scatter_add128<<<gridE32, 256, 0, stream>>>(bufA, src, dst, agg, E);
        wmma_gemm<true, false, true, 128, 128><<<gridRowT, 256, 0, stream>>>(
            agg, inv, Wl[L], bufA, Wr[L], bl[L], bufB, N);
        hipMemsetAsync(stats, 0, 2 * HID * 4, stream);
        colstats128<<<1024, 128, 0, stream>>>(bufB, N, stats);
        graphnorm_relu<<<gridEl, 256, 0, stream>>>(bufB, stats, gma[L], bta[L], alp[L], bufA, N);
    }

    // ---- MLP heads (chunked rows) ----
    float* out = (float*)d_out;
    for (int h = 0; h < 5; h++) {
        for (int c0 = 0; c0 < N; c0 += MLPCH) {
            int rows = (N - c0) < MLPCH ? (N - c0) : MLPCH;
            int gtile = (rows + 15) / 16;
            wmma_gemm<false, true, false, 128, 256><<<gtile, 256, 0, stream>>>(
                bufA + (size_t)c0 * HID, nullptr, W1[h], nullptr, nullptr, b1[h], t1, rows);
            wmma_gemm<false, true, false, 256, 128><<<gtile, 256, 0, stream>>>(
                t1, nullptr, W2[h], nullptr, nullptr, b2[h], t2, rows);
            long long nout = (long long)rows * oc[h];
            mlp_final<<<(int)((nout + 255) / 256), 256, 0, stream>>>(
                t2, W3[h], b3[h], out + headoff[h] + (size_t)c0 * oc[h], rows, oc[h]);
        }
    }
}